// SecureGNN_9517647528115
// MI455X (gfx1250) — compile-verified
//
#include <hip/hip_runtime.h>

typedef __attribute__((ext_vector_type(2))) float v2f;
typedef __attribute__((ext_vector_type(8))) float v8f;
typedef __attribute__((ext_vector_type(4))) int   v4i;

typedef __attribute__((address_space(1))) v4i gv4i;   // global-memory v4i
typedef __attribute__((address_space(3))) v4i lv4i;   // LDS v4i

#define DIM 128
#define NLAYERS 3
#define LDS_STRIDE 132   // 128 + 4 pad -> conflict-free 4m+kk bank pattern

#if __has_builtin(__builtin_amdgcn_global_load_async_to_lds_b128)
#define HAVE_ASYNC_LDS 1
#endif

// 16-byte global -> LDS copy, async (ASYNCcnt) when the toolchain has the
// gfx1250 builtin, plain register round-trip otherwise.
__device__ __forceinline__ void copy16_to_lds(const float* __restrict__ g,
                                              float* __restrict__ l) {
#ifdef HAVE_ASYNC_LDS
    // AS1 pointer from 64-bit address; AS3 pointer from the low 32 bits of
    // the generic LDS address.
    gv4i* gp = (gv4i*)(unsigned long long)(size_t)g;
    lv4i* lp = (lv4i*)(unsigned)(size_t)l;
    __builtin_amdgcn_global_load_async_to_lds_b128(gp, lp, 0, 0);
#else
    *(float4*)l = *(const float4*)g;
#endif
}

__device__ __forceinline__ void wait_async_lds() {
#ifdef HAVE_ASYNC_LDS
#if __has_builtin(__builtin_amdgcn_s_wait_asynccnt)
    __builtin_amdgcn_s_wait_asynccnt(0);
#else
    asm volatile("s_wait_asynccnt 0x0" ::: "memory");
#endif
#endif
}

// ---------------------------------------------------------------------------
// Fused triple GEMM: Xd = relu?(Xin)@Wd + bd ; Xc = ...@Wc + bc ;
// Acc = ...@Ws + bs  (Acc later receives the edge scatter-add).
// Block = 256 threads = 8 waves = 8 column tiles of one 16-row stripe.
// The A stripe (16x128, 8 KB) is staged once per block into LDS via the
// async global->LDS path and shared by all 8 waves; each wave then runs
// 32 K-steps of fp32 WMMA 16x16x4 with 3 accumulators.
// ---------------------------------------------------------------------------
__global__ __launch_bounds__(256) void gnn_gemm3_kernel(
    const float* __restrict__ Xin, int apply_relu,
    const float* __restrict__ Wd, const float* __restrict__ bd,
    const float* __restrict__ Wc, const float* __restrict__ bc,
    const float* __restrict__ Ws, const float* __restrict__ bs,
    float* __restrict__ Xd, float* __restrict__ Xc, float* __restrict__ Acc)
{
    __shared__ float sA[16 * LDS_STRIDE];

    const int lane    = threadIdx.x & 31;
    const int wave    = threadIdx.x >> 5;       // column tile 0..7
    const int rowTile = blockIdx.x;             // 16-row stripe
    const int colBase = wave * 16;

    // ---- stage A stripe: 16 rows x 128 cols, 8 floats per thread ----
    {
        const int r   = threadIdx.x >> 4;       // 0..15
        const int seg = (threadIdx.x & 15) * 8; // 0,8,...,120
        const float* g = Xin + ((size_t)rowTile * 16 + r) * DIM + seg;
        float* l       = &sA[r * LDS_STRIDE + seg];
        copy16_to_lds(g, l);
        copy16_to_lds(g + 4, l + 4);
    }
    wait_async_lds();
    __syncthreads();

    const int mrow  = lane & 15;                // A: row within tile
    const int khalf = (lane >> 4) << 1;         // A/B: K offset 0 or 2

    v8f accD = {}; v8f accC = {}; v8f accS = {};

    #pragma unroll 4
    for (int k0 = 0; k0 < DIM; k0 += 4) {
        const int kk = k0 + khalf;
        // A 16x4 from LDS: lanes 0-15 K=k0..k0+1, lanes 16-31 K=k0+2..k0+3
        const float2 av = *(const float2*)&sA[mrow * LDS_STRIDE + kk];
        v2f a;
        a.x = av.x;
        a.y = av.y;
        if (apply_relu) {
            a.x = fmaxf(a.x, 0.0f);
            a.y = fmaxf(a.y, 0.0f);
        }
        // B 4x16: same lane-half split over K, column = lane&15
        const int col = colBase + mrow;
        v2f bD, bC, bS;
        bD.x = Wd[(size_t)kk * DIM + col];  bD.y = Wd[(size_t)(kk + 1) * DIM + col];
        bC.x = Wc[(size_t)kk * DIM + col];  bC.y = Wc[(size_t)(kk + 1) * DIM + col];
        bS.x = Ws[(size_t)kk * DIM + col];  bS.y = Ws[(size_t)(kk + 1) * DIM + col];

        accD = __builtin_amdgcn_wmma_f32_16x16x4_f32(false, a, false, bD, (short)0, accD, false, false);
        accC = __builtin_amdgcn_wmma_f32_16x16x4_f32(false, a, false, bC, (short)0, accC, false, false);
        accS = __builtin_amdgcn_wmma_f32_16x16x4_f32(false, a, false, bS, (short)0, accS, false, false);
    }

    // C/D layout: VGPR i -> row i (lanes 0-15) / row i+8 (lanes 16-31)
    const int colOut = colBase + (lane & 15);
    const int rowOff = (lane >> 4) << 3;
    const float biasD = bd[colOut];
    const float biasC = bc[colOut];
    const float biasS = bs[colOut];

    #pragma unroll
    for (int i = 0; i < 8; ++i) {
        const size_t idx = ((size_t)rowTile * 16 + i + rowOff) * DIM + colOut;
        Xd[idx]  = accD[i] + biasD;
        Xc[idx]  = accC[i] + biasC;
        Acc[idx] = accS[i] + biasS;
    }
}

// ---------------------------------------------------------------------------
// Edge scatter: one wave per edge. msg = (type==0 ? Xd : Xc)[src];
// Acc[dst] += msg  via hardware fp32 atomics (4 floats per lane).
// ---------------------------------------------------------------------------
__global__ __launch_bounds__(256) void gnn_scatter_kernel(
    const float* __restrict__ Xd, const float* __restrict__ Xc,
    float* __restrict__ Acc,
    const int* __restrict__ src, const int* __restrict__ dst,
    const int* __restrict__ etype, int E)
{
    const int e = (int)((blockIdx.x * blockDim.x + threadIdx.x) >> 5);
    if (e >= E) return;
    const int lane = threadIdx.x & 31;

    const int s = src[e];
    const int d = dst[e];
    const int t = etype[e];

    const float* msg = (t == 0 ? Xd : Xc) + (size_t)s * DIM + lane * 4;
    const float4 v = *(const float4*)msg;

    float* out = Acc + (size_t)d * DIM + lane * 4;
    unsafeAtomicAdd(out + 0, v.x);
    unsafeAtomicAdd(out + 1, v.y);
    unsafeAtomicAdd(out + 2, v.z);
    unsafeAtomicAdd(out + 3, v.w);
}

// ---------------------------------------------------------------------------
// Global mean pool: one wave per node, ReLU applied inline (final activation),
// fp32 atomics into per-graph sums + counts.
// ---------------------------------------------------------------------------
__global__ __launch_bounds__(256) void gnn_pool_kernel(
    const float* __restrict__ Xf, const int* __restrict__ batch,
    float* __restrict__ sums, float* __restrict__ cnts, int N)
{
    const int n = (int)((blockIdx.x * blockDim.x + threadIdx.x) >> 5);
    if (n >= N) return;
    const int lane = threadIdx.x & 31;

    const int g = batch[n];
    const float4 v = *(const float4*)(Xf + (size_t)n * DIM + lane * 4);

    float* out = sums + (size_t)g * DIM + lane * 4;
    unsafeAtomicAdd(out + 0, fmaxf(v.x, 0.0f));
    unsafeAtomicAdd(out + 1, fmaxf(v.y, 0.0f));
    unsafeAtomicAdd(out + 2, fmaxf(v.z, 0.0f));
    unsafeAtomicAdd(out + 3, fmaxf(v.w, 0.0f));
    if (lane == 0) unsafeAtomicAdd(cnts + g, 1.0f);
}

__global__ void gnn_zero_kernel(float* __restrict__ p, int n)
{
    const int i = blockIdx.x * blockDim.x + threadIdx.x;
    if (i < n) p[i] = 0.0f;
}

__global__ void gnn_div_kernel(const float* __restrict__ sums,
                               const float* __restrict__ cnts,
                               float* __restrict__ out, int n)
{
    const int i = blockIdx.x * blockDim.x + threadIdx.x;
    if (i < n) out[i] = sums[i] / fmaxf(cnts[i >> 7], 1.0f);
}

// ---------------------------------------------------------------------------
// Launch. Input order: X, Wd, bd, Wc, bc, Ws, bs, edge_index, edge_types,
// batch_ids. Workspace: Xd | Xc | acc0 | acc1 | sums | cnts (~103 MB).
// ---------------------------------------------------------------------------
extern "C" void kernel_launch(void* const* d_in, const int* in_sizes, int n_in,
                              void* d_out, int out_size, void* d_ws, size_t ws_size,
                              hipStream_t stream) {
    const float* X  = (const float*)d_in[0];
    const float* Wd = (const float*)d_in[1];
    const float* bd = (const float*)d_in[2];
    const float* Wc = (const float*)d_in[3];
    const float* bc = (const float*)d_in[4];
    const float* Ws = (const float*)d_in[5];
    const float* bs = (const float*)d_in[6];
    const int* edge_index = (const int*)d_in[7];   // [2,E]
    const int* etype      = (const int*)d_in[8];   // [E]
    const int* batch      = (const int*)d_in[9];   // [N]

    const int N = in_sizes[0] / DIM;
    const int E = in_sizes[8];
    const int G = out_size / DIM;

    float* ws = (float*)d_ws;
    const size_t nd = (size_t)N * DIM;
    float* bufXd = ws;
    float* bufXc = ws + nd;
    float* acc0  = ws + 2 * nd;
    float* acc1  = ws + 3 * nd;
    float* sums  = ws + 4 * nd;
    float* cnts  = sums + (size_t)G * DIM;

    // zero pooling scratch (ws is poisoned by harness)
    const int zn = G * DIM + G;
    gnn_zero_kernel<<<(zn + 255) / 256, 256, 0, stream>>>(sums, zn);

    const int gemmBlocks    = (N + 15) / 16;                       // 3125
    const int scatterBlocks = (int)(((long long)E * 32 + 255) / 256);
    const int poolBlocks    = (int)(((long long)N * 32 + 255) / 256);

    const float* cur = X;
    for (int l = 0; l < NLAYERS; ++l) {
        float* acc = (l & 1) ? acc1 : acc0;
        gnn_gemm3_kernel<<<gemmBlocks, 256, 0, stream>>>(
            cur, (l > 0) ? 1 : 0,
            Wd + (size_t)l * DIM * DIM, bd + (size_t)l * DIM,
            Wc + (size_t)l * DIM * DIM, bc + (size_t)l * DIM,
            Ws + (size_t)l * DIM * DIM, bs + (size_t)l * DIM,
            bufXd, bufXc, acc);
        gnn_scatter_kernel<<<scatterBlocks, 256, 0, stream>>>(
            bufXd, bufXc, acc, edge_index, edge_index + E, etype, E);
        cur = acc;
    }

    gnn_pool_kernel<<<poolBlocks, 256, 0, stream>>>(cur, batch, sums, cnts, N);
    gnn_div_kernel<<<(G * DIM + 255) / 256, 256, 0, stream>>>(sums, cnts,
                                                              (float*)d_out, G * DIM);
}